// GCNLayer_29437705847356
// MI455X (gfx1250) — compile-verified
//
#include <hip/hip_runtime.h>
#include <hip/hip_bf16.h>

// ---------------------------------------------------------------------------
// GCN layer on MI455X (gfx1250, wave32).
// Phase 1: deterministic segment-sum (sorted rows -> binary search, no atomics)
// Phase 2: exact-fp32 channel GEMMs via V_WMMA_F32_16X16X4_F32
// Phase 3: masked segment-max pool with implicit 0-clamp
// ---------------------------------------------------------------------------

typedef __attribute__((ext_vector_type(2))) float v2f;
typedef __attribute__((ext_vector_type(8))) float v8f;

static __device__ __forceinline__ int lower_bound_i32(const int* __restrict__ a,
                                                      int n, int key) {
    int lo = 0, hi = n;
    while (lo < hi) {
        int mid = (lo + hi) >> 1;
        if (a[mid] < key) lo = mid + 1; else hi = mid;
    }
    return lo;
}

static __device__ __forceinline__ v8f wmma_f32_k4(v2f a, v2f b, v8f c) {
    // D = A(16x4, f32) * B(4x16, f32) + C(16x16, f32)  -- exact fp32 path
    return __builtin_amdgcn_wmma_f32_16x16x4_f32(
        /*neg_a=*/false, a, /*neg_b=*/false, b,
        /*c_mod=*/(short)0, c, /*reuse_a=*/false, /*reuse_b=*/false);
}

// ---------------------------------------------------------------------------
// Phase 1: agg[(b*N+n)*16 + i] = sum_{e in [lo_n,hi_n)} adj[e]*x[(b*N+col[e])*16+i]
// One block per node n; 128 threads = (b in 0..7) x (i in 0..15).
// ---------------------------------------------------------------------------
__global__ __launch_bounds__(128) void agg_kernel(
    const float* __restrict__ x, const float* __restrict__ adj,
    const int* __restrict__ row, const int* __restrict__ col,
    float* __restrict__ agg, int E, int N)
{
    __shared__ int s_range[2];
    const int n = blockIdx.x;
    const int t = threadIdx.x;
    if (t < 2) s_range[t] = lower_bound_i32(row, E, n + t);
    __syncthreads();
    const int lo = s_range[0], hi = s_range[1];
    const int b = t >> 4;        // 0..7
    const int i = t & 15;        // 0..15
    float acc = 0.0f;
    for (int e = lo; e < hi; ++e) {
        const int k = col[e];
        acc += adj[e] * x[((size_t)b * N + k) * 16 + i];
    }
    agg[((size_t)b * N + n) * 16 + i] = acc;
}

// ---------------------------------------------------------------------------
// Phase 2: h[r*64 + 0:32]  = agg[r,:] @ w_lin^T + b_lin
//          h[r*64 + 32:64] = x[r,:]   @ w_eye^T + b_eye
// for r = b*N+n in [0, 8*N). One wave (32 lanes) per 16-row tile.
//
// f32 WMMA layouts (ISA 7.12.2):
//   A 16x4 : lane m (0-15) holds A[m, {0,1}] in v[0],v[1];
//            lane 16+m holds A[m, {2,3}]   -> contiguous float2 per lane.
//   B 4x16 : lane m (0-15) holds B[{0,1}, m]; lane 16+m holds B[{2,3}, m].
//            B[k, o] = w[o*16 + k]          -> contiguous float2 per lane.
//   C/D    : vgpr v, lanes 0-15 -> (M=v,   N=lane),
//                    lanes 16-31 -> (M=v+8, N=lane-16).
// ---------------------------------------------------------------------------
__global__ __launch_bounds__(32) void gemm_h_kernel(
    const float* __restrict__ agg, const float* __restrict__ x,
    const float* __restrict__ w_lin, const float* __restrict__ b_lin,
    const float* __restrict__ w_eye, const float* __restrict__ b_eye,
    float* __restrict__ h)
{
    const int lane = threadIdx.x;           // 0..31, full wave active
    const int r0   = blockIdx.x * 16;       // first row of this M-tile
    const int m    = lane & 15;             // row/col-in-tile index
    const int kh   = (lane >> 4) << 1;      // 0 (lanes 0-15) or 2 (lanes 16-31)

    // Weight B-tiles: [matrix 0=lin,1=eye][o-tile 0,1][k-slice 0..3]
    v2f bw[2][2][4];
    const float* __restrict__ W[2] = { w_lin, w_eye };
#pragma unroll
    for (int wm = 0; wm < 2; ++wm)
#pragma unroll
        for (int ot = 0; ot < 2; ++ot)
#pragma unroll
            for (int kk = 0; kk < 4; ++kk)
                bw[wm][ot][kk] =
                    *(const v2f*)(W[wm] + (ot * 16 + m) * 16 + kk * 4 + kh);

    // A-tiles for this 16-row block (4 K-slices each)
    v2f aLin[4], aEye[4];
#pragma unroll
    for (int kk = 0; kk < 4; ++kk) {
        aLin[kk] = *(const v2f*)(agg + (size_t)(r0 + m) * 16 + kk * 4 + kh);
        aEye[kk] = *(const v2f*)(x   + (size_t)(r0 + m) * 16 + kk * 4 + kh);
    }

    // Accumulators seeded with broadcast bias (bias depends only on column).
    v8f cc0, cc1, cc2, cc3;
    {
        const float bl0 = b_lin[m], bl1 = b_lin[16 + m];
        const float be0 = b_eye[m], be1 = b_eye[16 + m];
#pragma unroll
        for (int v = 0; v < 8; ++v) {
            cc0[v] = bl0; cc1[v] = bl1; cc2[v] = be0; cc3[v] = be1;
        }
    }

#pragma unroll
    for (int kk = 0; kk < 4; ++kk) {
        cc0 = wmma_f32_k4(aLin[kk], bw[0][0][kk], cc0);   // lin cols  0..15
        cc1 = wmma_f32_k4(aLin[kk], bw[0][1][kk], cc1);   // lin cols 16..31
        cc2 = wmma_f32_k4(aEye[kk], bw[1][0][kk], cc2);   // eye cols 32..47
        cc3 = wmma_f32_k4(aEye[kk], bw[1][1][kk], cc3);   // eye cols 48..63
    }

    // Scatter D tiles into h[row*64 + col] (concat [lin | eye]).
    const int rbase = r0 + ((lane >> 4) << 3);
#pragma unroll
    for (int v = 0; v < 8; ++v) {
        const size_t rr = (size_t)(rbase + v) * 64;
        h[rr +  0 + m] = cc0[v];
        h[rr + 16 + m] = cc1[v];
        h[rr + 32 + m] = cc2[v];
        h[rr + 48 + m] = cc3[v];
    }
}

// ---------------------------------------------------------------------------
// Phase 3: out[b, j, c] = max(0, max_{e in [lo_j,hi_j)} adj[e]*h[(b*N+col[e])*64+c])
// for j in [0, N/2). One block per output node; 256 threads, each handles
// (b, c) and (b+4, c). Accumulator init 0 == the reference's max(seg, 0)
// clamp (deg < n always) and the empty-segment case.
// ---------------------------------------------------------------------------
__global__ __launch_bounds__(256) void pool_kernel(
    const float* __restrict__ h, const float* __restrict__ adj,
    const int* __restrict__ row, const int* __restrict__ col,
    float* __restrict__ out, int E, int N, int Nout)
{
    __shared__ int s_range[2];
    const int j = blockIdx.x;
    const int t = threadIdx.x;
    if (t < 2) s_range[t] = lower_bound_i32(row, E, j + t);
    __syncthreads();
    const int lo = s_range[0], hi = s_range[1];
    const int c  = t & 63;       // channel 0..63
    const int b0 = t >> 6;       // 0..3
    const int b1 = b0 + 4;
    float m0 = 0.0f, m1 = 0.0f;
    for (int e = lo; e < hi; ++e) {
        const int   k = col[e];
        const float v = adj[e];
        m0 = fmaxf(m0, v * h[((size_t)b0 * N + k) * 64 + c]);
        m1 = fmaxf(m1, v * h[((size_t)b1 * N + k) * 64 + c]);
    }
    out[((size_t)b0 * Nout + j) * 64 + c] = m0;
    out[((size_t)b1 * Nout + j) * 64 + c] = m1;
}

// ---------------------------------------------------------------------------
extern "C" void kernel_launch(void* const* d_in, const int* in_sizes, int n_in,
                              void* d_out, int out_size, void* d_ws, size_t ws_size,
                              hipStream_t stream) {
    const float* x     = (const float*)d_in[0];   // [B, N, 16]
    const float* w_lin = (const float*)d_in[1];   // [32, 16]
    const float* b_lin = (const float*)d_in[2];   // [32]
    const float* w_eye = (const float*)d_in[3];   // [32, 16]
    const float* b_eye = (const float*)d_in[4];   // [32]
    const float* adj   = (const float*)d_in[5];   // [E]
    const int*   row   = (const int*)  d_in[6];   // [E] sorted
    const int*   col   = (const int*)  d_in[7];   // [E]

    const int B  = 8;
    const int IN = 16;
    const int E  = in_sizes[5];
    const int N  = in_sizes[0] / (B * IN);        // 10000
    const int Nout = N / 2;                       // 5000

    float* agg = (float*)d_ws;                    // [B*N, 16]  (5.12 MB)
    float* h   = agg + (size_t)B * N * IN;        // [B*N, 64]  (20.5 MB)

    agg_kernel<<<N, 128, 0, stream>>>(x, adj, row, col, agg, E, N);

    const int tiles = (B * N) / 16;               // 5000 waves
    gemm_h_kernel<<<tiles, 32, 0, stream>>>(agg, x, w_lin, b_lin,
                                            w_eye, b_eye, h);

    pool_kernel<<<Nout, 256, 0, stream>>>(h, adj, row, col,
                                          (float*)d_out, E, N, Nout);
}